// FCOS_7851200217334
// MI455X (gfx1250) — compile-verified
//
#include <hip/hip_runtime.h>
#include <hip/hip_bf16.h>
#include <stdint.h>

#define IOU_THR 0.5f

typedef float v2f __attribute__((ext_vector_type(2)));
typedef float v8f __attribute__((ext_vector_type(8)));
typedef int   v2i __attribute__((ext_vector_type(2)));

typedef __attribute__((address_space(1))) v2i* as1_v2i_p;   // global int2*
typedef __attribute__((address_space(3))) v2i* as3_v2i_p;   // LDS int2*

// ---------------- kernel 1: decode deltas -> boxes, global max via atomicMax ----
__global__ void fcos_decode(const float* __restrict__ deltas,
                            const float* __restrict__ locs,
                            const int*   __restrict__ stride_p,
                            float* __restrict__ boxes,
                            int*   __restrict__ maxslot, int n)
{
    int i = blockIdx.x * blockDim.x + threadIdx.x;
    if (i >= n) return;
    float s  = (float)stride_p[0];
    float d0 = deltas[4*i+0], d1 = deltas[4*i+1];
    float d2 = deltas[4*i+2], d3 = deltas[4*i+3];
    bool bg = (d0 == -1.0f);
    float l = fmaxf(d0*s, 0.f), t = fmaxf(d1*s, 0.f);
    float r = fmaxf(d2*s, 0.f), b = fmaxf(d3*s, 0.f);
    float xc = locs[2*i], yc = locs[2*i+1];
    float x1 = bg ? xc : xc - l;
    float y1 = bg ? yc : yc - t;
    float x2 = bg ? xc : xc + r;
    float y2 = bg ? yc : yc + b;
    boxes[4*i+0]=x1; boxes[4*i+1]=y1; boxes[4*i+2]=x2; boxes[4*i+3]=y2;
    float mx = fmaxf(fmaxf(fmaxf(x1,y1), fmaxf(x2,y2)), 0.f); // global max is >= 0 (x2 >= loc >= 0)
    atomicMax(maxslot, __float_as_int(mx));
}

// ---------------- kernel 2: bitonic sort of (score, ~idx) keys, descending ------
__global__ void __launch_bounds__(1024)
fcos_sort(const float* __restrict__ scores, int* __restrict__ order, int n)
{
    __shared__ uint64_t key[8192];   // 64 KB LDS (gfx1250 WGP has 320 KB)
    const int t = threadIdx.x;
    for (int i = t; i < n; i += 1024) {
        uint32_t sb = __float_as_uint(scores[i]);          // scores in [0,1): bit-order == value-order
        key[i] = ((uint64_t)sb << 32) | (uint32_t)(~i);    // ~i => ties resolve to smaller idx first
    }
    __syncthreads();
    for (int k = 2; k <= n; k <<= 1) {
        for (int j = k >> 1; j > 0; j >>= 1) {
            for (int i = t; i < n; i += 1024) {
                int ij = i ^ j;
                if (ij > i) {
                    bool up = ((i & k) != 0);              // final pass => descending
                    uint64_t a = key[i], b = key[ij];
                    if ((a > b) == up) { key[i] = b; key[ij] = a; }
                }
            }
            __syncthreads();
        }
    }
    for (int i = t; i < n; i += 1024)
        order[i] = (int)(~(uint32_t)key[i]);
}

// ---------------- kernel 3: sorted, class-offset, clamped boxes + areas ---------
__global__ void fcos_prep(const float* __restrict__ boxes,
                          const int*   __restrict__ cls,
                          const int*   __restrict__ maxslot,
                          const int*   __restrict__ order,
                          float* __restrict__ sbox,
                          float* __restrict__ sarea, int n)
{
    int p = blockIdx.x * blockDim.x + threadIdx.x;
    if (p >= n) return;
    float maxc = __int_as_float(maxslot[0]);
    int i = order[p];
    float off = (float)cls[i] * (maxc + 1.0f);
    float x0 = fmaxf(boxes[4*i+0] + off, 0.f);
    float y0 = fmaxf(boxes[4*i+1] + off, 0.f);
    float x1 = fmaxf(boxes[4*i+2] + off, 0.f);
    float y1 = fmaxf(boxes[4*i+3] + off, 0.f);
    sbox[4*p+0]=x0; sbox[4*p+1]=y0; sbox[4*p+2]=x1; sbox[4*p+3]=y1;
    sarea[p] = (x1 - x0) * (y1 - y0);
}

// ---------------- kernel 4: 16x32 IoU bitmask tile per wave (WMMA union base) ---
__global__ void __launch_bounds__(32)
fcos_iou_mask(const float* __restrict__ sbox, const float* __restrict__ sarea,
              uint32_t* __restrict__ mask, int n)
{
    const int L    = threadIdx.x;        // wave32, all lanes active (WMMA needs EXEC all-1)
    const int half = L >> 4;
    const int nn   = L & 15;
    const int R0   = blockIdx.y * 16;    // 16 rows per block
    const int C0   = blockIdx.x * 32;    // 32 cols (one mask word) per block
    const int W    = n >> 5;

    __shared__ float rowBox[16 * 4];     // 16 row boxes staged in LDS

#if __has_builtin(__builtin_amdgcn_global_load_async_to_lds_b64)
    // CDNA5 async copy: each lane moves 8B, 256B total -> LDS, tracked by ASYNCcnt
    __builtin_amdgcn_global_load_async_to_lds_b64(
        (as1_v2i_p)(const_cast<float*>(sbox) + (size_t)R0 * 4 + L * 2),
        (as3_v2i_p)(&rowBox[L * 2]), 0, 0);
#if __has_builtin(__builtin_amdgcn_s_wait_asynccnt)
    __builtin_amdgcn_s_wait_asynccnt(0);
#else
    asm volatile("s_wait_asynccnt 0" ::: "memory");
#endif
    asm volatile("" ::: "memory");
#else
    rowBox[2*L]   = sbox[(size_t)R0*4 + 2*L];
    rowBox[2*L+1] = sbox[(size_t)R0*4 + 2*L + 1];
#endif
    __syncthreads();

    // Load unconditionally (addresses valid for all lanes), select afterwards —
    // avoids s_and_saveexec/branch around the loads.
    const float rowArea = sarea[R0 + nn];
    const float loSel   = (half == 0) ? 1.f : 0.f;

    // A (16x4 f32): lanes 0-15 hold K=0,1 = (area_m, 1); lanes 16-31 hold K=2,3 = 0
    v2f a;
    a.x = rowArea * loSel;   // exact: area >= 0 finite
    a.y = loSel;

    uint32_t bal0[8], bal1[8];
#pragma unroll
    for (int tile = 0; tile < 2; ++tile) {
        const int CB = C0 + tile * 16;
        const float colArea = sarea[CB + nn];
        // B (4x16 f32): lanes 0-15 hold K=0,1 = (1, area_n); lanes 16-31 hold K=2,3 = 0
        v2f b;
        b.x = loSel;
        b.y = colArea * loSel;
        v8f c = {};
        // s[r] = area_m + area_n for (m = r + half*8, n = nn) — rank-2 outer sum on the matrix core
        v8f s = __builtin_amdgcn_wmma_f32_16x16x4_f32(
                    false, a, false, b, (short)0, c, false, false);

        const float* cb = sbox + (size_t)(CB + nn) * 4;
        float cx0 = cb[0], cy0 = cb[1], cx1 = cb[2], cy1 = cb[3];
#pragma unroll
        for (int r = 0; r < 8; ++r) {
            int m = r + half * 8;
            float rx0 = rowBox[4*m+0], ry0 = rowBox[4*m+1];
            float rx1 = rowBox[4*m+2], ry1 = rowBox[4*m+3];
            float iw    = fmaxf(fminf(rx1, cx1) - fmaxf(rx0, cx0), 0.f);
            float ih    = fmaxf(fminf(ry1, cy1) - fmaxf(ry0, cy0), 0.f);
            float inter = iw * ih;
            float uni   = s[r] - inter;             // (area_m + area_n) - inter
            // iou > thr  <=>  inter > thr*uni  (uni >= 0; uni==0 -> false, same as NaN-compare)
            bool  pred  = inter > IOU_THR * uni;    // division-free: mul + cmp instead of IEEE div
            uint64_t bm = __ballot(pred);           // wave32: low16 = row r, high16 = row r+8
            if (tile == 0) bal0[r] = (uint32_t)bm; else bal1[r] = (uint32_t)bm;
        }
    }
#pragma unroll
    for (int r = 0; r < 8; ++r) {
        uint32_t wlo = (bal0[r] & 0xFFFFu) | (bal1[r] << 16);
        uint32_t whi = (bal0[r] >> 16)     | (bal1[r] & 0xFFFF0000u);
        if (L == r)     mask[(size_t)(R0 + r)     * W + blockIdx.x] = wlo;
        if (L == 8 + r) mask[(size_t)(R0 + 8 + r) * W + blockIdx.x] = whi;
    }
}

// ---------------- kernel 5: serial greedy scan over bitmask rows ---------------
__global__ void __launch_bounds__(256)
fcos_scan(const uint32_t* __restrict__ mask, const int* __restrict__ order,
          float* __restrict__ keep, int n)
{
    const int t = threadIdx.x;
    const int W = n >> 5;                // 256 words per row == blockDim
    for (int i = t; i < n; i += 256) keep[i] = 0.f;
    __shared__ int alive_s;
    __syncthreads();
    uint32_t remv = 0;                   // thread t owns suppression word t
    for (int p = 0; p < n; ++p) {
        uint32_t row = mask[(size_t)p * W + t];
        if (p + 1 < n) __builtin_prefetch(&mask[(size_t)(p + 1) * W + t], 0, 0);
        if (t == (p >> 5)) alive_s = (int)(((remv >> (p & 31)) & 1u) ^ 1u);
        __syncthreads();
        int alive = alive_s;
        __syncthreads();
        if (alive) {
            remv |= row;
            if (t == 0) keep[order[p]] = 1.0f;
        }
    }
}

extern "C" void kernel_launch(void* const* d_in, const int* in_sizes, int n_in,
                              void* d_out, int out_size, void* d_ws, size_t ws_size,
                              hipStream_t stream)
{
    const float* deltas = (const float*)d_in[0];
    const float* locs   = (const float*)d_in[1];
    const float* scores = (const float*)d_in[2];
    const int*   cls    = (const int*)d_in[3];
    const int*   stride = (const int*)d_in[4];
    const int n = in_sizes[0] / 4;       // 8192

    float* outBoxes = (float*)d_out;
    float* outKeep  = outBoxes + (size_t)n * 4;

    char* ws = (char*)d_ws;
    int*      maxslot = (int*)ws;                                        // 64 B
    int*      order   = (int*)(ws + 64);                                 // n*4
    float*    sbox    = (float*)(ws + 64 + (size_t)n * 4);               // n*16
    float*    sarea   = (float*)(ws + 64 + (size_t)n * 20);              // n*4
    uint32_t* mask    = (uint32_t*)(ws + 64 + (size_t)n * 24);           // n*(n/32)*4 = 8 MB

    (void)hipMemsetAsync(maxslot, 0, 64, stream);
    fcos_decode<<<(n + 255) / 256, 256, 0, stream>>>(deltas, locs, stride, outBoxes, maxslot, n);
    fcos_sort  <<<1, 1024, 0, stream>>>(scores, order, n);
    fcos_prep  <<<(n + 255) / 256, 256, 0, stream>>>(outBoxes, cls, maxslot, order, sbox, sarea, n);
    dim3 g(n / 32, n / 16);
    fcos_iou_mask<<<g, 32, 0, stream>>>(sbox, sarea, mask, n);
    fcos_scan  <<<1, 256, 0, stream>>>(mask, order, outKeep, n);
}